// head_2327872274482
// MI455X (gfx1250) — compile-verified
//
#include <hip/hip_runtime.h>

typedef __attribute__((ext_vector_type(2))) float v2f;
typedef __attribute__((ext_vector_type(8))) float v8f;

#define B_SZ  1024
#define T_SZ  256
#define C_SZ  195
#define H_SZ  8
// scores scaled by 1/sqrt(C) per reference
#define SCALE_QK (0.0716114874f)   // 1/sqrt(195)

static __device__ __forceinline__ v8f wmma_f32_4(v2f a, v2f b, v8f c) {
  // D = A(16x4 f32) * B(4x16 f32) + C(16x16 f32)
  return __builtin_amdgcn_wmma_f32_16x16x4_f32(
      /*neg_a=*/false, a, /*neg_b=*/false, b,
      /*c_mod=*/(short)0, c, /*reuse_a=*/false, /*reuse_b=*/false);
}

__global__ __launch_bounds__(256) void attn_head_kernel(
    const float* __restrict__ x,   // [B,T,C]
    const float* __restrict__ Wq,  // [C,H]
    const float* __restrict__ Wk,  // [C,H]
    const float* __restrict__ Wv,  // [C,H]
    float* __restrict__ out)       // [B,T,H]
{
  // ---- LDS ----
  __shared__ float s_wqk[196 * 16];   // [c][n]: n<8 -> Wq, n>=8 -> Wk; c=195 zero pad
  __shared__ float s_wv [196 * 16];   // [c][n]: n<8 -> Wv, else 0
  __shared__ float s_q[T_SZ * H_SZ];
  __shared__ float s_k[T_SZ * H_SZ];
  __shared__ float s_v[T_SZ * H_SZ];
  __shared__ float s_p[8][16 * 16];   // per-wave P tile bounce buffer

  const int b    = blockIdx.x;
  const int tid  = threadIdx.x;
  const int lane = tid & 31;
  const int wave = tid >> 5;
  const int half = lane >> 4;     // 0: lanes 0-15, 1: lanes 16-31
  const int l16  = lane & 15;
  const int koff = half * 2;      // K offset for A/B operand lane layout

  // ---- stage packed weights into LDS (zero-padded to c=196) ----
  for (int idx = tid; idx < 196 * 16; idx += 256) {
    const int c = idx >> 4, n = idx & 15;
    float wqk = 0.0f, wv = 0.0f;
    if (c < C_SZ) {
      wqk = (n < H_SZ) ? Wq[c * H_SZ + n] : Wk[c * H_SZ + (n - 8)];
      if (n < H_SZ) wv = Wv[c * H_SZ + n];
    }
    s_wqk[idx] = wqk;
    s_wv [idx] = wv;
  }
  __syncthreads();

  const float* xb = x + (size_t)b * T_SZ * C_SZ;

  // ================= Phase 1: q,k,v = x @ [Wq|Wk], x @ [Wv|0] =================
  // wave owns t-rows [32w, 32w+32): two 16-row A-tiles sharing each B operand.
  {
    const int t0 = wave * 32;
    const float* xrow0 = xb + (size_t)(t0      + l16) * C_SZ;  // A-tile 0 row
    const float* xrow1 = xb + (size_t)(t0 + 16 + l16) * C_SZ;  // A-tile 1 row

    v8f acc_qk0 = {}, acc_v0 = {};
    v8f acc_qk1 = {}, acc_v1 = {};

    // 48 full K=4 steps (c = 0..191): 2 LDS B-pairs feed 4 independent WMMA chains
    for (int c0 = 0; c0 < 192; c0 += 4) {
      const int c = c0 + koff;
      v2f a0;  a0[0] = xrow0[c];  a0[1] = xrow0[c + 1];
      v2f a1;  a1[0] = xrow1[c];  a1[1] = xrow1[c + 1];
      v2f bqk; bqk[0] = s_wqk[c * 16 + l16]; bqk[1] = s_wqk[(c + 1) * 16 + l16];
      v2f bv;  bv[0]  = s_wv [c * 16 + l16]; bv[1]  = s_wv [(c + 1) * 16 + l16];
      acc_qk0 = wmma_f32_4(a0, bqk, acc_qk0);
      acc_qk1 = wmma_f32_4(a1, bqk, acc_qk1);
      acc_v0  = wmma_f32_4(a0, bv,  acc_v0);
      acc_v1  = wmma_f32_4(a1, bv,  acc_v1);
    }
    // tail step c0 = 192 (c = 192..194, pad c=195 with zero)
    {
      const int c = 192 + koff;
      v2f a0;
      a0[0] = xrow0[c];
      a0[1] = (c + 1 < C_SZ) ? xrow0[c + 1] : 0.0f;   // guard OOB at c=195
      v2f a1;
      a1[0] = xrow1[c];
      a1[1] = (c + 1 < C_SZ) ? xrow1[c + 1] : 0.0f;
      v2f bqk; bqk[0] = s_wqk[c * 16 + l16]; bqk[1] = s_wqk[(c + 1) * 16 + l16];
      v2f bv;  bv[0]  = s_wv [c * 16 + l16]; bv[1]  = s_wv [(c + 1) * 16 + l16];
      acc_qk0 = wmma_f32_4(a0, bqk, acc_qk0);
      acc_qk1 = wmma_f32_4(a1, bqk, acc_qk1);
      acc_v0  = wmma_f32_4(a0, bv,  acc_v0);
      acc_v1  = wmma_f32_4(a1, bv,  acc_v1);
    }

    // scatter D tiles (row M=r in VGPR r; lanes 16-31 hold M=r+8; N = l16)
#pragma unroll
    for (int r = 0; r < 8; ++r) {
      const int ta = t0 + r + half * 8;
      const int tb = ta + 16;
      if (l16 < H_SZ) {
        s_q[ta * H_SZ + l16] = acc_qk0[r];
        s_v[ta * H_SZ + l16] = acc_v0[r];
        s_q[tb * H_SZ + l16] = acc_qk1[r];
        s_v[tb * H_SZ + l16] = acc_v1[r];
      } else {
        s_k[ta * H_SZ + (l16 - 8)] = acc_qk0[r];
        s_k[tb * H_SZ + (l16 - 8)] = acc_qk1[r];
      }
    }
  }
  __syncthreads();

  // ================= Phase 2: causal flash attention =================
  // balance the causal triangle: wave w takes q-tiles {w, 15-w}
  float* ptile = &s_p[wave][0];

  for (int tile = 0; tile < 2; ++tile) {
    const int qt = (tile == 0) ? wave : (15 - wave);
    const int q0 = qt * 16;

    // Q A-operands for this tile (K = 0..3 and 4..7)
    const int qrow = q0 + l16;
    v2f aq0; aq0[0] = s_q[qrow * H_SZ + koff];     aq0[1] = s_q[qrow * H_SZ + koff + 1];
    v2f aq1; aq1[0] = s_q[qrow * H_SZ + 4 + koff]; aq1[1] = s_q[qrow * H_SZ + 4 + koff + 1];

    float m_run[8], l_run[8];
#pragma unroll
    for (int r = 0; r < 8; ++r) { m_run[r] = -__builtin_inff(); l_run[r] = 0.0f; }
    v8f o = {};

    for (int j0 = 0; j0 <= q0; j0 += 16) {
      // S tile = Q(16x8) @ K^T(8x16) : two K=4 WMMAs
      v2f bk0; bk0[0] = s_k[(j0 + l16) * H_SZ + koff];     bk0[1] = s_k[(j0 + l16) * H_SZ + koff + 1];
      v2f bk1; bk1[0] = s_k[(j0 + l16) * H_SZ + 4 + koff]; bk1[1] = s_k[(j0 + l16) * H_SZ + 4 + koff + 1];
      v8f s = {};
      s = wmma_f32_4(aq0, bk0, s);
      s = wmma_f32_4(aq1, bk1, s);

      const bool diag = (j0 == q0);

      // online softmax: D layout keeps row stats in exactly the right lanes
#pragma unroll
      for (int r = 0; r < 8; ++r) {
        const int m = r + half * 8;           // row within tile
        float sv = s[r] * SCALE_QK;
        if (diag && (l16 > m)) sv = -__builtin_inff();   // causal mask on diagonal tile

        float mx = sv;                         // row max across 16 lanes of half-wave
        mx = fmaxf(mx, __shfl_xor(mx, 1));
        mx = fmaxf(mx, __shfl_xor(mx, 2));
        mx = fmaxf(mx, __shfl_xor(mx, 4));
        mx = fmaxf(mx, __shfl_xor(mx, 8));

        const float mnew  = fmaxf(m_run[r], mx);
        const float alpha = __expf(m_run[r] - mnew);
        const float pv    = __expf(sv - mnew);

        float sum = pv;                        // row sum across 16 lanes of half-wave
        sum += __shfl_xor(sum, 1);
        sum += __shfl_xor(sum, 2);
        sum += __shfl_xor(sum, 4);
        sum += __shfl_xor(sum, 8);

        l_run[r] = l_run[r] * alpha + sum;
        m_run[r] = mnew;
        o[r] *= alpha;

        ptile[m * 16 + l16] = pv;              // bounce P through LDS for A-layout
      }

      // O += P(16x16) @ V(16x8 padded to 16) : four K=4 WMMAs
#pragma unroll
      for (int ck = 0; ck < 4; ++ck) {
        const int kk = ck * 4 + koff;
        v2f ap; ap[0] = ptile[l16 * 16 + kk]; ap[1] = ptile[l16 * 16 + kk + 1];
        v2f bv;
        bv[0] = (l16 < H_SZ) ? s_v[(j0 + kk)     * H_SZ + l16] : 0.0f;
        bv[1] = (l16 < H_SZ) ? s_v[(j0 + kk + 1) * H_SZ + l16] : 0.0f;
        o = wmma_f32_4(ap, bv, o);
      }
    }

    // finalize: divide by row sum, store out[b, q0+m, h]
#pragma unroll
    for (int r = 0; r < 8; ++r) {
      const int t = q0 + r + half * 8;
      const float val = o[r] / l_run[r];
      if (l16 < H_SZ) out[((size_t)b * T_SZ + t) * H_SZ + l16] = val;
    }
  }
}

extern "C" void kernel_launch(void* const* d_in, const int* in_sizes, int n_in,
                              void* d_out, int out_size, void* d_ws, size_t ws_size,
                              hipStream_t stream) {
  const float* x  = (const float*)d_in[0];
  const float* Wq = (const float*)d_in[1];
  const float* Wk = (const float*)d_in[2];
  const float* Wv = (const float*)d_in[3];
  float* out = (float*)d_out;
  (void)in_sizes; (void)n_in; (void)out_size; (void)d_ws; (void)ws_size;

  attn_head_kernel<<<dim3(B_SZ), dim3(256), 0, stream>>>(x, Wq, Wk, Wv, out);
}